// GATRegressor_32856499814415
// MI455X (gfx1250) — compile-verified
//
#include <hip/hip_runtime.h>
#include <hip/hip_bf16.h>
#include <math.h>

// ---------------- problem constants (match reference) ----------------
static constexpr int NN   = 50000;
static constexpr int EE   = 1600000;
static constexpr int GG   = 100;
static constexpr int INC  = 16;
static constexpr int H1   = 4,  C1 = 32;   // GAT1: heads x per-head ch (concat -> 128)
static constexpr int H2   = 1,  C2 = 32;   // GAT2
static constexpr int HC1  = H1 * C1;       // 128
static constexpr int HC2  = H2 * C2;       // 32
static constexpr float NEG_SLOPE = 0.2f;

typedef float    v2f  __attribute__((ext_vector_type(2)));
typedef float    v8f  __attribute__((ext_vector_type(8)));
typedef unsigned u32x4 __attribute__((ext_vector_type(4)));
typedef int      i32x4 __attribute__((ext_vector_type(4)));
typedef int      i32x8 __attribute__((ext_vector_type(8)));

// ---------------- helpers ----------------
__device__ __forceinline__ float lrelu(float x) { return x > 0.f ? x : NEG_SLOPE * x; }
__device__ __forceinline__ float elu1(float x)  { return x > 0.f ? x : (expf(x) - 1.f); }

// order-preserving float -> uint key (for atomicMax-based segment max)
__device__ __forceinline__ unsigned f2key(float f) {
    unsigned i = __float_as_uint(f);
    return (i & 0x80000000u) ? ~i : (i | 0x80000000u);
}
__device__ __forceinline__ float key2f(unsigned u) {
    unsigned i = (u & 0x80000000u) ? (u & 0x7FFFFFFFu) : ~u;
    return __uint_as_float(i);
}
static constexpr unsigned NEGINF_KEY = 0x007FFFFFu; // f2key(-inf)

// ---------------- generic fills ----------------
__global__ void k_zero_f32(float* p, long long n) {
    long long i = (long long)blockIdx.x * blockDim.x + threadIdx.x;
    if (i < n) p[i] = 0.f;
}
__global__ void k_fill_u32(unsigned* p, long long n, unsigned v) {
    long long i = (long long)blockIdx.x * blockDim.x + threadIdx.x;
    if (i < n) p[i] = v;
}

// ---------------------------------------------------------------------------
// fp32 WMMA GEMM: C[M,NCOL] = A[M,K] @ B[K,NCOL]   (K, NCOL multiples of 16/4)
// 256 threads = 8 waves per block; block covers 128 rows x all NCOL columns.
// B (8-16 KB) is staged into LDS once per block via the Tensor Data Mover
// (tensor_load_to_lds + s_wait_tensorcnt), then each wave runs
// V_WMMA_F32_16X16X4_F32 with one A fragment reused across NCOL/16 col tiles.
// ---------------------------------------------------------------------------
template<int K, int NCOL>
__global__ void k_gemm_tiled(const float* __restrict__ A,
                             const float* __restrict__ B,
                             float* __restrict__ C, int M) {
    constexpr int NT = NCOL / 16;
    __shared__ float Bs[K * NCOL];

    const int tid  = threadIdx.x;
    const int wave = tid >> 5;
    const int lane = tid & 31;
    const int m16  = lane & 15;
    const int half = lane >> 4;

#if __has_builtin(__builtin_amdgcn_tensor_load_to_lds) && __has_builtin(__builtin_amdgcn_s_wait_tensorcnt)
    // --- TDM: DMA the whole B matrix (K x NCOL f32, contiguous) into LDS ---
    if (wave == 0) {
        unsigned long long ga = (unsigned long long)(uintptr_t)B;
        unsigned ldsb = (unsigned)(uintptr_t)(&Bs[0]);   // flat low 32 bits == LDS byte offset
        // D# group 0: count=1 | lds_addr | global_addr[56:0] | type=2 (ISA 8.3)
        u32x4 g0;
        g0.x = 1u;
        g0.y = ldsb;
        g0.z = (unsigned)ga;
        g0.w = (unsigned)((ga >> 32) & 0x1FFFFFFull) | (2u << 30);
        // D# group 1 (ISA 8.4): data_size=4B; 2-D tensor NCOL x K; tile = whole tensor
        i32x8 g1;
        g1[0] = (int)(2u << 16);                              // data_size = 2 (4 bytes)
        g1[1] = (int)(((unsigned)NCOL & 0xFFFFu) << 16);      // tensor_dim0[15:0] @ bits 63:48
        g1[2] = (int)((((unsigned)NCOL >> 16) & 0xFFFFu) |
                      (((unsigned)K & 0xFFFFu) << 16));       // tensor_dim0 hi | tensor_dim1 lo
        g1[3] = (int)((((unsigned)K >> 16) & 0xFFFFu) |
                      (((unsigned)NCOL & 0xFFFFu) << 16));    // tensor_dim1 hi | tile_dim0
        g1[4] = (int)((unsigned)K & 0xFFFFu);                 // tile_dim1 (tile_dim2 unused)
        g1[5] = (int)(unsigned)NCOL;                          // tensor_dim0_stride[31:0]
        g1[6] = 0;                                            // stride0 hi | stride1 lo
        g1[7] = 0;
        i32x4 gz  = {0, 0, 0, 0};                             // groups 2/3 unused (2-D)
        i32x8 gz8 = {0, 0, 0, 0, 0, 0, 0, 0};                 // trailing group (6-arg form)
        __builtin_amdgcn_tensor_load_to_lds(g0, g1, gz, gz, gz8, 0);
        __builtin_amdgcn_s_wait_tensorcnt((short)0);
    }
    __syncthreads();
#else
    for (int i = tid; i < K * NCOL; i += 256) Bs[i] = B[i];
    __syncthreads();
#endif

    const int  rowA = blockIdx.x * 128 + wave * 16 + m16;     // A row this lane feeds
    const bool rok  = rowA < M;
    const float* Arow = A + (long long)rowA * K;

    v8f zero = {};
    v8f acc[NT];
    #pragma unroll
    for (int t = 0; t < NT; ++t) acc[t] = zero;

    for (int k = 0; k < K; k += 4) {
        const int kq = k + half * 2;          // K interleaved across half-waves
        v2f a;
        a.x = rok ? Arow[kq + 0] : 0.f;
        a.y = rok ? Arow[kq + 1] : 0.f;
        #pragma unroll
        for (int t = 0; t < NT; ++t) {
            v2f b;
            b.x = Bs[(kq + 0) * NCOL + t * 16 + m16];
            b.y = Bs[(kq + 1) * NCOL + t * 16 + m16];
            acc[t] = __builtin_amdgcn_wmma_f32_16x16x4_f32(
                false, a, false, b, (short)0, acc[t], false, false);
        }
    }

    // C/D 16x16 f32 layout: VGPR v -> row = v + 8*half, col = m16
    const int r0 = blockIdx.x * 128 + wave * 16 + 8 * half;
    #pragma unroll
    for (int v = 0; v < 8; ++v) {
        const int r = r0 + v;
        if (r < M) {
            #pragma unroll
            for (int t = 0; t < NT; ++t)
                C[(long long)r * NCOL + t * 16 + m16] = acc[t][v];
        }
    }
}

// ---------------- per-node attention logits a_s/a_d [N,H] ----------------
__global__ void k_attn_logits(const float* __restrict__ hfeat,   // [N, H*C]
                              const float* __restrict__ att_s,   // [H, C]
                              const float* __restrict__ att_d,   // [H, C]
                              float* __restrict__ as_, float* __restrict__ ad_,
                              int n, int H, int C) {
    int idx = blockIdx.x * blockDim.x + threadIdx.x;   // over n*H
    if (idx >= n * H) return;
    int node = idx / H, h = idx - node * H;
    const float* hp = hfeat + (long long)node * H * C + h * C;
    float s = 0.f, d = 0.f;
    for (int c = 0; c < C; ++c) {
        float v = hp[c];
        s += v * att_s[h * C + c];
        d += v * att_d[h * C + c];
    }
    as_[idx] = s;
    ad_[idx] = d;
}

// ---------------- edge pass 1: segment max of leaky-relu logits -------------
__global__ void k_edge_max(const int* __restrict__ ei, int E, int n,
                           const float* __restrict__ as_, const float* __restrict__ ad_,
                           unsigned* __restrict__ mkey, int H) {
    long long idx = (long long)blockIdx.x * blockDim.x + threadIdx.x; // over (E+n)*H
    long long tot = (long long)(E + n) * H;
    if (idx >= tot) return;
    long long e = idx / H;
    int h = (int)(idx - e * H);
    int s, d;
    if (e < E) { s = ei[e]; d = ei[(long long)E + e]; } else { s = d = (int)(e - E); }
    if (e + 8192 < E) {                      // stream-ahead: global_prefetch_b8
        __builtin_prefetch(&ei[e + 8192], 0, 1);
        __builtin_prefetch(&ei[(long long)E + e + 8192], 0, 1);
    }
    float logit = lrelu(as_[(long long)s * H + h] + ad_[(long long)d * H + h]);
    atomicMax(&mkey[(long long)d * H + h], f2key(logit));
}

// ---------------- edge pass 2: softmax denominators -------------------------
__global__ void k_edge_denom(const int* __restrict__ ei, int E, int n,
                             const float* __restrict__ as_, const float* __restrict__ ad_,
                             const unsigned* __restrict__ mkey,
                             float* __restrict__ den, int H) {
    long long idx = (long long)blockIdx.x * blockDim.x + threadIdx.x;
    long long tot = (long long)(E + n) * H;
    if (idx >= tot) return;
    long long e = idx / H;
    int h = (int)(idx - e * H);
    int s, d;
    if (e < E) { s = ei[e]; d = ei[(long long)E + e]; } else { s = d = (int)(e - E); }
    if (e + 8192 < E) {
        __builtin_prefetch(&ei[e + 8192], 0, 1);
        __builtin_prefetch(&ei[(long long)E + e + 8192], 0, 1);
    }
    float logit = lrelu(as_[(long long)s * H + h] + ad_[(long long)d * H + h]);
    float m = key2f(mkey[(long long)d * H + h]);
    atomicAdd(&den[(long long)d * H + h], expf(logit - m));
}

// ---------------- edge pass 3: weighted scatter-aggregate -------------------
// one block of H*C threads per edge; channel index == threadIdx.x (coalesced)
__global__ void k_edge_agg(const int* __restrict__ ei, int E, int n,
                           const float* __restrict__ as_, const float* __restrict__ ad_,
                           const unsigned* __restrict__ mkey, const float* __restrict__ den,
                           const float* __restrict__ hfeat,   // [N, H*C]
                           float* __restrict__ agg,           // [N, H*C]
                           int H, int C) {
    long long e = blockIdx.x;
    int tid = threadIdx.x;          // 0 .. H*C-1
    int h = tid / C;
    int s, d;
    if (e < E) { s = ei[e]; d = ei[(long long)E + e]; } else { s = d = (int)(e - E); }
    if (tid == 0 && e + 2048 < E) {
        __builtin_prefetch(&ei[e + 2048], 0, 1);
        __builtin_prefetch(&ei[(long long)E + e + 2048], 0, 1);
    }
    float logit = lrelu(as_[(long long)s * H + h] + ad_[(long long)d * H + h]);
    float m  = key2f(mkey[(long long)d * H + h]);
    float dn = den[(long long)d * H + h];
    float alpha = expf(logit - m) / (dn + 1e-16f);
    atomicAdd(&agg[(long long)d * H * C + tid],
              alpha * hfeat[(long long)s * H * C + tid]);
}

// ---------------- bias + ELU (in place) -------------------------------------
__global__ void k_bias_elu(float* __restrict__ agg, const float* __restrict__ bias,
                           int n, int HC) {
    long long i = (long long)blockIdx.x * blockDim.x + threadIdx.x;
    if (i >= (long long)n * HC) return;
    int j = (int)(i % HC);
    agg[i] = elu1(agg[i] + bias[j]);
}

// ---------------- global mean pool (accumulation part) ----------------------
__global__ void k_pool(const float* __restrict__ y, const int* __restrict__ batch,
                       float* __restrict__ sums, float* __restrict__ cnt, int n, int C) {
    long long i = (long long)blockIdx.x * blockDim.x + threadIdx.x;
    if (i >= (long long)n * C) return;
    int node = (int)(i / C), c = (int)(i - (long long)node * C);
    int g = batch[node];
    atomicAdd(&sums[(long long)g * C + c], y[i]);
    if (c == 0) atomicAdd(&cnt[g], 1.f);
}

// ---------------- MLP head: relu(pooled@Wh1+bh1)@Wh2+bh2 --------------------
__global__ void k_head(const float* __restrict__ sums, const float* __restrict__ cnt,
                       const float* __restrict__ Wh1, const float* __restrict__ bh1,
                       const float* __restrict__ Wh2, const float* __restrict__ bh2,
                       float* __restrict__ out, int G) {
    int g = blockIdx.x * blockDim.x + threadIdx.x;
    if (g >= G) return;
    float pooled[C2];
    float inv = 1.f / fmaxf(cnt[g], 1.f);
    #pragma unroll
    for (int c = 0; c < C2; ++c) pooled[c] = sums[g * C2 + c] * inv;
    float acc = 0.f;
    for (int j = 0; j < 64; ++j) {
        float hj = bh1[j];
        #pragma unroll
        for (int c = 0; c < C2; ++c) hj += pooled[c] * Wh1[c * 64 + j];
        hj = hj > 0.f ? hj : 0.f;
        acc += hj * Wh2[j];
    }
    out[g] = acc + bh2[0];
}

// ---------------- launch ----------------------------------------------------
static inline long long cdiv(long long a, long long b) { return (a + b - 1) / b; }

extern "C" void kernel_launch(void* const* d_in, const int* in_sizes, int n_in,
                              void* d_out, int out_size, void* d_ws, size_t ws_size,
                              hipStream_t stream) {
    const float* x    = (const float*)d_in[0];
    const int*   ei   = (const int*)d_in[1];     // [2,E]
    const int*   bat  = (const int*)d_in[2];     // [N]
    const float* W1   = (const float*)d_in[3];
    const float* as1w = (const float*)d_in[4];
    const float* ad1w = (const float*)d_in[5];
    const float* b1   = (const float*)d_in[6];
    const float* W2   = (const float*)d_in[7];
    const float* as2w = (const float*)d_in[8];
    const float* ad2w = (const float*)d_in[9];
    const float* b2   = (const float*)d_in[10];
    const float* Wh1  = (const float*)d_in[11];
    const float* bh1  = (const float*)d_in[12];
    const float* Wh2  = (const float*)d_in[13];
    const float* bh2  = (const float*)d_in[14];
    float* out = (float*)d_out;

    const int n = NN, E = EE, G = GG;
    const long long Etot = (long long)E + n;

    // ---- carve workspace (bytes), 256B aligned chunks ----
    char* w = (char*)d_ws;
    size_t off = 0;
    auto carve = [&](size_t bytes) { void* p = w + off; off = (off + bytes + 255) & ~(size_t)255; return p; };
    float*    h1lin = (float*)   carve((size_t)n * HC1 * 4);   // x@W1
    float*    agg1  = (float*)   carve((size_t)n * HC1 * 4);   // GAT1 aggregate -> y1 (in place)
    float*    h2lin = (float*)   carve((size_t)n * HC2 * 4);   // y1@W2
    float*    agg2  = (float*)   carve((size_t)n * HC2 * 4);   // GAT2 aggregate -> y2 (in place)
    float*    as1   = (float*)   carve((size_t)n * H1 * 4);
    float*    ad1   = (float*)   carve((size_t)n * H1 * 4);
    unsigned* mk1   = (unsigned*)carve((size_t)n * H1 * 4);
    float*    den1  = (float*)   carve((size_t)n * H1 * 4);
    float*    as2   = (float*)   carve((size_t)n * H2 * 4);
    float*    ad2   = (float*)   carve((size_t)n * H2 * 4);
    unsigned* mk2   = (unsigned*)carve((size_t)n * H2 * 4);
    float*    den2  = (float*)   carve((size_t)n * H2 * 4);
    float*    sums  = (float*)   carve((size_t)G * C2 * 4);
    float*    cnt   = (float*)   carve((size_t)G * 4);
    (void)ws_size; (void)n_in; (void)in_sizes; (void)out_size;

    const int TB = 256;

    // ---- re-init all accumulators every call (harness does not re-poison) ----
    k_zero_f32<<<cdiv((long long)n * HC1, TB), TB, 0, stream>>>(agg1, (long long)n * HC1);
    k_zero_f32<<<cdiv((long long)n * HC2, TB), TB, 0, stream>>>(agg2, (long long)n * HC2);
    k_zero_f32<<<cdiv((long long)n * H1, TB), TB, 0, stream>>>(den1, (long long)n * H1);
    k_zero_f32<<<cdiv((long long)n * H2, TB), TB, 0, stream>>>(den2, (long long)n * H2);
    k_zero_f32<<<cdiv((long long)G * C2, TB), TB, 0, stream>>>(sums, (long long)G * C2);
    k_zero_f32<<<1, TB, 0, stream>>>(cnt, G);
    k_fill_u32<<<cdiv((long long)n * H1, TB), TB, 0, stream>>>(mk1, (long long)n * H1, NEGINF_KEY);
    k_fill_u32<<<cdiv((long long)n * H2, TB), TB, 0, stream>>>(mk2, (long long)n * H2, NEGINF_KEY);

    // ---- GAT layer 1 ----
    k_gemm_tiled<INC, HC1><<<cdiv(n, 128), 256, 0, stream>>>(x, W1, h1lin, n);
    k_attn_logits<<<cdiv((long long)n * H1, TB), TB, 0, stream>>>(h1lin, as1w, ad1w, as1, ad1, n, H1, C1);
    k_edge_max  <<<cdiv(Etot * H1, TB), TB, 0, stream>>>(ei, E, n, as1, ad1, mk1, H1);
    k_edge_denom<<<cdiv(Etot * H1, TB), TB, 0, stream>>>(ei, E, n, as1, ad1, mk1, den1, H1);
    k_edge_agg  <<<(unsigned)Etot, HC1, 0, stream>>>(ei, E, n, as1, ad1, mk1, den1, h1lin, agg1, H1, C1);
    k_bias_elu  <<<cdiv((long long)n * HC1, TB), TB, 0, stream>>>(agg1, b1, n, HC1);   // agg1 == y1

    // ---- GAT layer 2 ----
    k_gemm_tiled<HC1, HC2><<<cdiv(n, 128), 256, 0, stream>>>(agg1, W2, h2lin, n);
    k_attn_logits<<<cdiv((long long)n * H2, TB), TB, 0, stream>>>(h2lin, as2w, ad2w, as2, ad2, n, H2, C2);
    k_edge_max  <<<cdiv(Etot * H2, TB), TB, 0, stream>>>(ei, E, n, as2, ad2, mk2, H2);
    k_edge_denom<<<cdiv(Etot * H2, TB), TB, 0, stream>>>(ei, E, n, as2, ad2, mk2, den2, H2);
    k_edge_agg  <<<(unsigned)Etot, HC2, 0, stream>>>(ei, E, n, as2, ad2, mk2, den2, h2lin, agg2, H2, C2);
    k_bias_elu  <<<cdiv((long long)n * HC2, TB), TB, 0, stream>>>(agg2, b2, n, HC2);   // agg2 == y2

    // ---- pool + head ----
    k_pool<<<cdiv((long long)n * C2, TB), TB, 0, stream>>>(agg2, bat, sums, cnt, n, C2);
    k_head<<<1, 128, 0, stream>>>(sums, cnt, Wh1, bh1, Wh2, bh2, out, G);
}